// ClassicalReconstructionHydraSSMCore_64527588655205
// MI455X (gfx1250) — compile-verified
//
#include <hip/hip_runtime.h>
#include <math.h>

#define NQ     10
#define DIM    1024        // 2^10 amplitudes
#define NPAIR  512
#define BLOCK  256         // 8 wave32
#define NP     100         // params per circuit

typedef float v2f __attribute__((ext_vector_type(2)));
typedef float v8f __attribute__((ext_vector_type(8)));

struct cplx { float re, im; };
__device__ __forceinline__ cplx cmul(cplx a, cplx b) {
    return { a.re * b.re - a.im * b.im, a.re * b.im + a.im * b.re };
}

// insert a 0 bit at position t (pair index -> amplitude index with wire bit = 0)
__device__ __forceinline__ int pair_amp0(int p, int t) {
    return ((p >> t) << (t + 1)) | (p & ((1 << t) - 1));
}

// ---------------------------------------------------------------------------
// Apply a 2x2 complex gate G on `wire` via V_WMMA_F32_16X16X4_F32.
//
// Real embedding: v=(re0,im0,re1,im1) per amplitude pair, v' = M v with
//   M = [[ gr00,-gi00, gr01,-gi01],
//        [ gi00, gr00, gi01, gr01],
//        [ gr10,-gi10, gr11,-gi11],
//        [ gi10, gr10, gi11, gr11]]
// Mapping: A(16x4) holds M in rows 0..3 (rest zero); B(4x16) holds 16 pairs
// as columns (row k = component k).  D rows 0..3 = updated components, and
// for pair n they sit in lane n as (D[0],D[1],D[2],D[3]) -> two b64 stores.
// Pairs are partitioned across the 8 waves (wave wv owns [wv*64, wv*64+64)),
// so no intra-gate cross-wave hazards; caller barriers between gates.
// ---------------------------------------------------------------------------
__device__ __forceinline__ void fused_1q_wmma(float2* psi, int wire,
                                              cplx g00, cplx g01,
                                              cplx g10, cplx g11, int tid)
{
    const int t    = 9 - wire;         // bit position of this wire
    const int lane = tid & 31;
    const int wv   = tid >> 5;         // wave id 0..7
    const int n    = lane & 15;
    const int hi   = lane >> 4;        // 0: K=0,1 / amp i0 ; 1: K=2,3 / amp i1

    // A operand: lane<16 -> (A[m][0], A[m][1]); lane>=16 -> (A[m][2], A[m][3])
    const cplx ga = hi ? g01 : g00;    // column-0/1 entries for K-halves
    const cplx gb = hi ? g11 : g10;
    float a0 = 0.0f, a1 = 0.0f;
    if (n < 4) {
        a0 = (n == 0) ?  ga.re : (n == 1) ? ga.im : (n == 2) ?  gb.re : gb.im;
        a1 = (n == 0) ? -ga.im : (n == 1) ? ga.re : (n == 2) ? -gb.im : gb.re;
    }
    v2f A = { a0, a1 };

#pragma unroll
    for (int j = 0; j < 4; ++j) {
        const int pairbase = (wv * 4 + j) * 16;
        // B operand: column n = pair pairbase+n; lane<16 supplies rows 0,1
        // = (re,im) of amp i0; lane>=16 supplies rows 2,3 = (re,im) of amp i1.
        const int p   = pairbase + n;
        const int i0  = pair_amp0(p, t);
        const int i1  = i0 | (1 << t);
        const int amp = hi ? i1 : i0;
        float2 bv = psi[amp];
        v2f B = { bv.x, bv.y };
        v8f C = {};
        v8f D = __builtin_amdgcn_wmma_f32_16x16x4_f32(
            false, A, false, B, (short)0, C, false, false);
        // Result for pair n lives in lane n (lanes 0-15), VGPRs 0..3.
        if (hi == 0) {
            psi[i0] = make_float2(D[0], D[1]);
            psi[i1] = make_float2(D[2], D[3]);
        }
    }
}

// CRX(theta, ctrl, tgt): RX on target where ctrl bit == 1 (256 pairs).
__device__ __forceinline__ void apply_crx(float2* psi, int ctrl, int tgt,
                                          float theta, int tid)
{
    const int tc = 9 - ctrl, tt = 9 - tgt;
    const int lo  = tc < tt ? tc : tt;
    const int hi2 = tc < tt ? tt : tc;
    int x = ((tid >> lo) << (lo + 1)) | (tid & ((1 << lo) - 1));
    x     = ((x >> hi2) << (hi2 + 1)) | (x & ((1 << hi2) - 1));
    const int i0 = x | (1 << tc);
    const int i1 = i0 | (1 << tt);
    const float h = 0.5f * theta;
    const float cc = __cosf(h), ss = __sinf(h);
    float2 a0 = psi[i0], a1 = psi[i1];
    // a0' = c*a0 - i*s*a1 ; a1' = -i*s*a0 + c*a1
    float2 r0 = { cc * a0.x + ss * a1.y, cc * a0.y - ss * a1.x };
    float2 r1 = { ss * a0.y + cc * a1.x, -ss * a0.x + cc * a1.y };
    psi[i0] = r0; psi[i1] = r1;
}

// Fused G = RZ(tz) * RY(ty) * RX(tx)
__device__ __forceinline__ void make_fused(float tx, float ty, float tz,
                                           cplx& g00, cplx& g01,
                                           cplx& g10, cplx& g11)
{
    const float h1 = 0.5f * tx, h2 = 0.5f * ty, h3 = 0.5f * tz;
    const float c1 = __cosf(h1), s1 = __sinf(h1);
    const float c2 = __cosf(h2), s2 = __sinf(h2);
    const float c3 = __cosf(h3), s3 = __sinf(h3);
    // M = RY*RX
    cplx m00 = {  c2 * c1,  s2 * s1 };
    cplx m01 = { -s2 * c1, -c2 * s1 };
    cplx m10 = {  s2 * c1, -c2 * s1 };
    cplx m11 = {  c2 * c1, -s2 * s1 };
    cplx e0 = { c3, -s3 }, e1 = { c3, s3 };
    g00 = cmul(e0, m00); g01 = cmul(e0, m01);
    g10 = cmul(e1, m10); g11 = cmul(e1, m11);
}

__device__ void run_circuit(float2* psi, const float* sP, int pbase,
                            float dtv, bool backward, int tid)
{
    // |0...0>
    for (int k = tid; k < DIM; k += BLOCK)
        psi[k] = make_float2(k == 0 ? 1.0f : 0.0f, 0.0f);
    __syncthreads();

    // RY input embedding (wires 0..9)
    for (int i = 0; i < NQ; ++i) {
        const float h = 0.5f * sP[i];
        const float c = __cosf(h), s = __sinf(h);
        cplx g00 = { c, 0.f }, g01 = { -s, 0.f }, g10 = { s, 0.f }, g11 = { c, 0.f };
        fused_1q_wmma(psi, i, g00, g01, g10, g11, tid);
        __syncthreads();
    }

    int off = pbase;
    for (int layer = 0; layer < 2; ++layer) {
        for (int k = 0; k < NQ; ++k) {
            const int i = backward ? (NQ - 1 - k) : k;
            cplx g00, g01, g10, g11;
            make_fused(sP[off] * dtv, sP[off + 1] * dtv, sP[off + 2] * dtv,
                       g00, g01, g10, g11);
            off += 3;
            fused_1q_wmma(psi, i, g00, g01, g10, g11, tid);
            __syncthreads();
        }
        if (!backward) {
            for (int i = 0; i < NQ; ++i)      { apply_crx(psi, i, (i + 1) % NQ, sP[off++], tid); __syncthreads(); }
            for (int i = NQ - 1; i >= 0; --i) { apply_crx(psi, i, (i + 9) % NQ, sP[off++], tid); __syncthreads(); }
        } else {
            for (int i = NQ - 1; i >= 0; --i) { apply_crx(psi, i, (i + 9) % NQ, sP[off++], tid); __syncthreads(); }
            for (int i = 0; i < NQ; ++i)      { apply_crx(psi, i, (i + 1) % NQ, sP[off++], tid); __syncthreads(); }
        }
    }
}

__global__ __launch_bounds__(BLOCK)
void ClassicalReconstructionHydraSSMCore_64527588655205_kernel(
    const float* __restrict__ angles, const float* __restrict__ fwd,
    const float* __restrict__ bwd,    const float* __restrict__ diag,
    const float* __restrict__ dts,
    const float* ar, const float* ai, const float* br, const float* bi,
    const float* gr, const float* gi,
    float* __restrict__ out)
{
    __shared__ float  sP[312];     // 0..9 angles | 10..109 fwd | 110..209 bwd | 210..309 diag | 310 dt
    __shared__ float2 psiW[DIM];   // working state
    __shared__ float2 psiA[DIM];   // accumulator
    __shared__ float  red0[BLOCK], red1[BLOCK], red2[BLOCK];

    const int tid = threadIdx.x;
    const int b   = blockIdx.x;

    // ---- stage this batch element's params into LDS via async global->LDS ----
    if (tid < NQ) {
        unsigned l   = (unsigned)(uintptr_t)&sP[tid];
        unsigned off = (unsigned)((b * NQ + tid) * 4);
        asm volatile("global_load_async_to_lds_b32 %0, %1, %2"
                     :: "v"(l), "v"(off), "s"(angles) : "memory");
    }
    if (tid < NP) {
        unsigned l1  = (unsigned)(uintptr_t)&sP[10 + tid];
        unsigned l2  = (unsigned)(uintptr_t)&sP[110 + tid];
        unsigned l3  = (unsigned)(uintptr_t)&sP[210 + tid];
        unsigned off = (unsigned)((b * NP + tid) * 4);
        asm volatile("global_load_async_to_lds_b32 %0, %1, %2"
                     :: "v"(l1), "v"(off), "s"(fwd) : "memory");
        asm volatile("global_load_async_to_lds_b32 %0, %1, %2"
                     :: "v"(l2), "v"(off), "s"(bwd) : "memory");
        asm volatile("global_load_async_to_lds_b32 %0, %1, %2"
                     :: "v"(l3), "v"(off), "s"(diag) : "memory");
    }
    if (tid == 0) {
        unsigned l   = (unsigned)(uintptr_t)&sP[310];
        unsigned off = (unsigned)(b * 4);
        asm volatile("global_load_async_to_lds_b32 %0, %1, %2"
                     :: "v"(l), "v"(off), "s"(dts) : "memory");
    }
    asm volatile("s_wait_asynccnt 0" ::: "memory");
    __syncthreads();

    // ---- complex combination weights ----
    const float arv = *ar, aiv = *ai, brv = *br, biv = *bi, grv = *gr, giv = *gi;
    const float nw  = sqrtf(arv*arv + aiv*aiv + brv*brv + biv*biv + grv*grv + giv*giv + 1e-9f);
    cplx W[3] = { { arv / nw, aiv / nw }, { brv / nw, biv / nw }, { grv / nw, giv / nw } };
    const float dtv = sP[310];

    // ---- three circuits: forward (dt), backward (dt), diagonal (dt=1) ----
    for (int c = 0; c < 3; ++c) {
        const int   pbase = 10 + c * NP;
        const bool  bwdf  = (c == 1);
        const float d     = (c == 2) ? 1.0f : dtv;
        run_circuit(psiW, sP, pbase, d, bwdf, tid);
        __syncthreads();
        const cplx w = W[c];
        for (int k = tid; k < DIM; k += BLOCK) {
            float2 v = psiW[k];
            float2 t2 = { w.re * v.x - w.im * v.y, w.re * v.y + w.im * v.x };
            if (c == 0) { psiA[k] = t2; }
            else        { psiA[k].x += t2.x; psiA[k].y += t2.y; }
        }
        __syncthreads();
    }

    // ---- L2 norm of combined state (deterministic tree reduction) ----
    {
        float s = 0.f;
        for (int k = tid; k < DIM; k += BLOCK) {
            float2 v = psiA[k];
            s += v.x * v.x + v.y * v.y;
        }
        red0[tid] = s;
        __syncthreads();
        for (int o = BLOCK / 2; o > 0; o >>= 1) {
            if (tid < o) red0[tid] += red0[tid + o];
            __syncthreads();
        }
    }
    const float vn    = sqrtf(red0[0]) + 1e-9f;
    const float scale = 1.0f / (vn * vn);   // fold normalization into quadratic observables
    __syncthreads();

    // ---- Pauli X/Y/Z expectations per wire ----
    for (int w = 0; w < NQ; ++w) {
        const int t = 9 - w;
        float xr = 0.f, xi = 0.f, zz = 0.f;
        for (int m = tid; m < NPAIR; m += BLOCK) {
            const int i0 = pair_amp0(m, t);
            const int i1 = i0 | (1 << t);
            float2 p0 = psiA[i0], p1 = psiA[i1];
            xr += p0.x * p1.x + p0.y * p1.y;           // Re(conj(p0)*p1)
            xi += p0.x * p1.y - p0.y * p1.x;           // Im(conj(p0)*p1)
            zz += (p0.x * p0.x + p0.y * p0.y) - (p1.x * p1.x + p1.y * p1.y);
        }
        red0[tid] = xr; red1[tid] = xi; red2[tid] = zz;
        __syncthreads();
        for (int o = BLOCK / 2; o > 0; o >>= 1) {
            if (tid < o) {
                red0[tid] += red0[tid + o];
                red1[tid] += red1[tid + o];
                red2[tid] += red2[tid + o];
            }
            __syncthreads();
        }
        if (tid == 0) {
            out[b * 30 + w]      = 2.0f * red0[0] * scale;
            out[b * 30 + 10 + w] = 2.0f * red1[0] * scale;
            out[b * 30 + 20 + w] = red2[0] * scale;
        }
        __syncthreads();
    }
}

extern "C" void kernel_launch(void* const* d_in, const int* in_sizes, int n_in,
                              void* d_out, int out_size, void* d_ws, size_t ws_size,
                              hipStream_t stream)
{
    (void)n_in; (void)out_size; (void)d_ws; (void)ws_size;
    const float* angles = (const float*)d_in[0];
    const float* fwd    = (const float*)d_in[1];
    const float* bwd    = (const float*)d_in[2];
    const float* diag   = (const float*)d_in[3];
    const float* dts    = (const float*)d_in[4];
    const float* ar     = (const float*)d_in[5];
    const float* ai     = (const float*)d_in[6];
    const float* br     = (const float*)d_in[7];
    const float* bi     = (const float*)d_in[8];
    const float* gr     = (const float*)d_in[9];
    const float* gi     = (const float*)d_in[10];
    float* out = (float*)d_out;

    const int batch = in_sizes[0] / NQ;   // 8192
    ClassicalReconstructionHydraSSMCore_64527588655205_kernel
        <<<batch, BLOCK, 0, stream>>>(angles, fwd, bwd, diag, dts,
                                      ar, ai, br, bi, gr, gi, out);
}